// TreeVariationalPosterior_23914377904202
// MI455X (gfx1250) — compile-verified
//
#include <hip/hip_runtime.h>
#include <stdint.h>

#define NUM_NODES 4096
#define EPS 1e-10f

// ---- order-preserving float <-> uint encoding for atomicMax over floats ----
__device__ __forceinline__ unsigned enc_f32(float f) {
    unsigned b = __float_as_uint(f);
    return (b & 0x80000000u) ? ~b : (b | 0x80000000u);
}
__device__ __forceinline__ float dec_f32(unsigned k) {
    unsigned b = (k & 0x80000000u) ? (k ^ 0x80000000u) : ~k;
    return __uint_as_float(b);
}

// ---- zero per-node accumulators (ws is never re-poisoned between replays) ----
__global__ void k_init(unsigned* segkey, float* segsum, float* S,
                       int* count, int* cursor, int N) {
    int n = blockIdx.x * blockDim.x + threadIdx.x;
    if (n < N) {
        segkey[n] = 0u;      // encoded -inf lower bound (all real keys are larger)
        segsum[n] = 0.0f;
        S[n]      = 0.0f;
        count[n]  = 0;
        cursor[n] = 0;
    }
}

// ---- per-edge: segment max (encoded) + child count ----
__global__ void k_max_count(const float* __restrict__ logits,
                            const int* __restrict__ parent,
                            unsigned* segkey, int* count, int E) {
    int e = blockIdx.x * blockDim.x + threadIdx.x;
    if (e < E) {
        int p = parent[e];
        atomicMax(&segkey[p], enc_f32(logits[e]));
        atomicAdd(&count[p], 1);
    }
}

// ---- per-edge: ex = exp(l - segmax), segment sum ----
__global__ void k_exp(const float* __restrict__ logits,
                      const int* __restrict__ parent,
                      const unsigned* __restrict__ segkey,
                      float* exb, float* segsum, int E) {
    int e = blockIdx.x * blockDim.x + threadIdx.x;
    if (e < E) {
        int p = parent[e];
        float m  = dec_f32(segkey[p]);
        float ex = expf(logits[e] - m);
        exb[e] = ex;
        atomicAdd(&segsum[p], ex);
    }
}

// ---- per-edge: B = ex / segsum, S[parent] += B ----
__global__ void k_B(const int* __restrict__ parent,
                    const float* __restrict__ exb,
                    const float* __restrict__ segsum,
                    float* B, float* S, int E) {
    int e = blockIdx.x * blockDim.x + threadIdx.x;
    if (e < E) {
        int p = parent[e];
        float b = exb[e] / segsum[p];
        B[e] = b;
        atomicAdd(&S[p], b);
    }
}

// ---- single-block exclusive prefix sum over counts -> CSR offsets ----
__global__ void k_scan(const int* __restrict__ count, int* off, int N) {
    __shared__ int part[1024];
    int t   = threadIdx.x;
    int per = (N + (int)blockDim.x - 1) / (int)blockDim.x;   // 4 for N=4096
    int base = t * per;
    int local[8];
    int s = 0;
    for (int j = 0; j < per && j < 8; j++) {
        int v = (base + j < N) ? count[base + j] : 0;
        local[j] = s;            // exclusive within chunk
        s += v;
    }
    part[t] = s;
    __syncthreads();
    for (int d = 1; d < (int)blockDim.x; d <<= 1) {
        int v = 0;
        if (t >= d) v = part[t - d];
        __syncthreads();
        if (t >= d) part[t] += v;
        __syncthreads();
    }
    int cb = (t == 0) ? 0 : part[t - 1];
    for (int j = 0; j < per && j < 8; j++)
        if (base + j < N) off[base + j] = cb + local[j];
    if (t == (int)blockDim.x - 1) off[N] = part[blockDim.x - 1];
}

// ---- bucket edge ids by parent ----
__global__ void k_csr(const int* __restrict__ parent,
                      const int* __restrict__ off,
                      int* cursor, int* csr, int E) {
    int e = blockIdx.x * blockDim.x + threadIdx.x;
    if (e < E) {
        int p = parent[e];
        int pos = atomicAdd(&cursor[p], 1);
        csr[off[p] + pos] = e;
    }
}

// ---- bulk zero-fill of A (256 MB) via CDNA5 async LDS->global b128 stores ----
__global__ void k_fill(float* __restrict__ A, size_t chunks16) {
    __shared__ float4 zbuf[256];
    zbuf[threadIdx.x] = make_float4(0.0f, 0.0f, 0.0f, 0.0f);
    __syncthreads();
    // low 32 bits of a flat shared pointer == LDS byte offset (gfx1250 aperture rule)
    unsigned lds = (unsigned)(uintptr_t)(void*)&zbuf[threadIdx.x];
    size_t i      = (size_t)blockIdx.x * blockDim.x + threadIdx.x;
    size_t stride = (size_t)gridDim.x * blockDim.x;
    unsigned long long base = (unsigned long long)(uintptr_t)A;
    for (; i < chunks16; i += stride) {
        unsigned long long g = base + (i << 4);
        // per-lane 16B DMA: copies 16 bytes from the LDS address in lds to the global address in g, tracked by ASYNCcnt
        asm volatile("global_store_async_from_lds_b128 %0, %1, off"
                     :: "v"(g), "v"(lds) : "memory");
    }
    asm volatile("s_wait_asynccnt 0" ::: "memory");
}

// ---- scatter nonzeros + diagonal (last-write-wins) + Z ----
__global__ void k_scatter(const int* __restrict__ parent,
                          const int* __restrict__ head,
                          const float* __restrict__ B,
                          const float* __restrict__ S,
                          const int* __restrict__ off,
                          const int* __restrict__ csr,
                          float* __restrict__ A,
                          float* __restrict__ Zout, int E) {
    int i = blockIdx.x * blockDim.x + threadIdx.x;
    if (i >= E) return;
    int h = head[i];
    float Zi = fmaxf(0.5f + S[h], EPS);
    Zout[i] = Zi;
    float invZ = 1.0f / Zi;
    size_t rowbase = (size_t)i * (size_t)E;
    int s = off[h], t = off[h + 1];
    for (int j = s; j < t; j++) {
        int k = csr[j];
        A[rowbase + k] = B[k] * invZ;
    }
    // diagonal written after scatter -> matches reference's .at[idx,idx].set override
    float dv = ((h == parent[i]) ? B[i] : 0.5f) * invZ;
    A[rowbase + i] = dv;
}

extern "C" void kernel_launch(void* const* d_in, const int* in_sizes, int n_in,
                              void* d_out, int out_size, void* d_ws, size_t ws_size,
                              hipStream_t stream) {
    const float* logits = (const float*)d_in[0];
    const int*   parent = (const int*)d_in[1];
    const int*   head   = (const int*)d_in[2];
    const int E = in_sizes[0];
    const int N = NUM_NODES;

    float* A    = (float*)d_out;
    float* Zout = A + (size_t)E * (size_t)E;

    // carve workspace (256B-aligned slices)
    char* w = (char*)d_ws;
    auto carve = [&](size_t bytes) {
        char* p = w;
        w += (bytes + 255) & ~(size_t)255;
        return p;
    };
    unsigned* segkey = (unsigned*)carve((size_t)N * 4);
    float*    segsum = (float*)   carve((size_t)N * 4);
    float*    S      = (float*)   carve((size_t)N * 4);
    int*      count  = (int*)     carve((size_t)N * 4);
    int*      cursor = (int*)     carve((size_t)N * 4);
    int*      off    = (int*)     carve((size_t)(N + 1) * 4);
    float*    exb    = (float*)   carve((size_t)E * 4);
    float*    B      = (float*)   carve((size_t)E * 4);
    int*      csr    = (int*)     carve((size_t)E * 4);

    const int T = 256;
    const int gE = (E + T - 1) / T;
    const int gN = (N + T - 1) / T;

    // The 256MB zero-fill is the long pole; everything else is noise.
    size_t chunks16 = ((size_t)E * (size_t)E * sizeof(float)) >> 4;
    k_fill<<<4096, T, 0, stream>>>(A, chunks16);

    k_init<<<gN, T, 0, stream>>>(segkey, segsum, S, count, cursor, N);
    k_max_count<<<gE, T, 0, stream>>>(logits, parent, segkey, count, E);
    k_exp<<<gE, T, 0, stream>>>(logits, parent, segkey, exb, segsum, E);
    k_B<<<gE, T, 0, stream>>>(parent, exb, segsum, B, S, E);
    k_scan<<<1, 1024, 0, stream>>>(count, off, N);
    k_csr<<<gE, T, 0, stream>>>(parent, off, cursor, csr, E);
    k_scatter<<<gE, T, 0, stream>>>(parent, head, B, S, off, csr, A, Zout, E);
}